// Net_88819923681343
// MI455X (gfx1250) — compile-verified
//
#include <hip/hip_runtime.h>
#include <cstdint>

#define TPB 256
#define SLOTS 2                  // float4 tiles per thread per step
#define TILE_F4 (TPB * SLOTS)    // 512 float4 = 1024 points per tile
#define TILE_PTS (TILE_F4 * 2)
#define MAX_BLOCKS 2048
#define NLAYERS 10

// vector type matching the builtin's parameter ('vector_size(4*sizeof(int)) int')
typedef int v4i __attribute__((vector_size(16)));
typedef __attribute__((address_space(1))) v4i gv4i;   // global
typedef __attribute__((address_space(3))) v4i lv4i;   // LDS

// ---------- CDNA5 helpers -------------------------------------------------

__device__ __forceinline__ float fast_tanh(float x) {
#if __has_builtin(__builtin_amdgcn_tanhf)
  return __builtin_amdgcn_tanhf(x);
#elif __has_builtin(__builtin_amdgcn_tanh_f32)
  return __builtin_amdgcn_tanh_f32(x);
#else
  float r;
  asm("v_tanh_f32 %0, %1" : "=v"(r) : "v"(x));
  return r;
#endif
}

__device__ __forceinline__ float bcast(float v) {
  // force weight into an SGPR (value is wave-uniform)
  return __uint_as_float(__builtin_amdgcn_readfirstlane(__float_as_uint(v)));
}

// async global -> LDS, 16B per lane, tracked by ASYNCcnt
__device__ __forceinline__ void async_load_f4(const float4* g, const float4* l) {
#if __has_builtin(__builtin_amdgcn_global_load_async_to_lds_b128)
  __builtin_amdgcn_global_load_async_to_lds_b128(
      (gv4i*)(uintptr_t)g,
      (lv4i*)(uint32_t)(uintptr_t)l,   // generic shared addr: low 32 bits = LDS byte offset
      0, 0);
#else
  uint64_t ga = (uint64_t)(uintptr_t)g;
  uint32_t la = (uint32_t)(uintptr_t)l;
  asm volatile("global_load_async_to_lds_b128 %0, %1, off"
               :: "v"(la), "v"(ga) : "memory");
#endif
}

template <int N>
__device__ __forceinline__ void wait_async() {
#if __has_builtin(__builtin_amdgcn_s_wait_asynccnt)
  __builtin_amdgcn_s_wait_asynccnt(N);
#else
  if constexpr (N == 0) asm volatile("s_wait_asynccnt 0x0" ::: "memory");
  else                  asm volatile("s_wait_asynccnt 0x2" ::: "memory");
#endif
}

// ---------- main tiled kernel --------------------------------------------

__global__ __launch_bounds__(TPB) void mlp_tiles(
    const float* __restrict__ x, const float* __restrict__ Ws,
    const float* __restrict__ bs, const float* __restrict__ Wout,
    const float* __restrict__ bout, float* __restrict__ out,
    int numTiles, int tilesPerBlock) {
  __shared__ float4 buf[2][TILE_F4];  // 16 KB double buffer
  const int tid = threadIdx.x;

  // weights -> SGPRs, fully unrolled layer chain keeps them scalar
  float W[NLAYERS][4], B[NLAYERS][2];
#pragma unroll
  for (int l = 0; l < NLAYERS; ++l) {
    W[l][0] = bcast(Ws[4 * l + 0]);
    W[l][1] = bcast(Ws[4 * l + 1]);
    W[l][2] = bcast(Ws[4 * l + 2]);
    W[l][3] = bcast(Ws[4 * l + 3]);
    B[l][0] = bcast(bs[2 * l + 0]);
    B[l][1] = bcast(bs[2 * l + 1]);
  }
  const float wo0 = bcast(Wout[0]), wo1 = bcast(Wout[1]), bo = bcast(bout[0]);

  const int t0 = blockIdx.x * tilesPerBlock;
  const int t1 = (t0 + tilesPerBlock < numTiles) ? (t0 + tilesPerBlock) : numTiles;
  if (t0 >= t1) return;

  const float4* x4 = (const float4*)x;
  float2* o2 = (float2*)out;

  // prefetch first tile (2 async ops per wave -> ASYNCcnt = 2)
  {
    const float4* g = x4 + (size_t)t0 * TILE_F4;
    async_load_f4(g + tid,       &buf[0][tid]);
    async_load_f4(g + TPB + tid, &buf[0][TPB + tid]);
  }

  for (int t = t0; t < t1; ++t) {
    const int cur = (t - t0) & 1;
    const bool more = (t + 1) < t1;
    if (more) {
      const float4* g = x4 + (size_t)(t + 1) * TILE_F4;
      async_load_f4(g + tid,       &buf[cur ^ 1][tid]);
      async_load_f4(g + TPB + tid, &buf[cur ^ 1][TPB + tid]);
      wait_async<2>();   // next tile's 2 ops may stay in flight; current tile done
    } else {
      wait_async<0>();
    }

    // each wave reads only the LDS slots it wrote -> no barriers needed
    const float4 p = buf[cur][tid];
    const float4 q = buf[cur][TPB + tid];

    // 4 independent points per thread to feed the trans (tanh) pipe
    float h0[4] = {p.x, p.z, q.x, q.z};
    float h1[4] = {p.y, p.w, q.y, q.w};
#pragma unroll
    for (int l = 0; l < NLAYERS; ++l) {
      float a0[4], a1[4];
#pragma unroll
      for (int k = 0; k < 4; ++k) {
        a0[k] = fmaf(W[l][0], h0[k], fmaf(W[l][1], h1[k], B[l][0]));
        a1[k] = fmaf(W[l][2], h0[k], fmaf(W[l][3], h1[k], B[l][1]));
      }
#pragma unroll
      for (int k = 0; k < 4; ++k) {
        h0[k] = fast_tanh(a0[k]);
        h1[k] = fast_tanh(a1[k]);
      }
    }
    float o[4];
#pragma unroll
    for (int k = 0; k < 4; ++k) o[k] = fmaf(wo0, h0[k], fmaf(wo1, h1[k], bo));

    // float4 index f holds points (2f, 2f+1) -> out pair index f (coalesced b64 stores)
    const size_t i0 = (size_t)t * TILE_F4 + tid;
    o2[i0]       = make_float2(o[0], o[1]);
    o2[i0 + TPB] = make_float2(o[2], o[3]);
  }
}

// ---------- tail (non-multiple-of-tile remainder; N=2^23 makes this empty) --

__global__ __launch_bounds__(TPB) void mlp_tail(
    const float* __restrict__ x, const float* __restrict__ Ws,
    const float* __restrict__ bs, const float* __restrict__ Wout,
    const float* __restrict__ bout, float* __restrict__ out,
    int start, int n) {
  int i = start + blockIdx.x * blockDim.x + threadIdx.x;
  if (i >= n) return;
  float h0 = x[2 * i], h1 = x[2 * i + 1];
#pragma unroll
  for (int l = 0; l < NLAYERS; ++l) {
    float a0 = fmaf(Ws[4 * l + 0], h0, fmaf(Ws[4 * l + 1], h1, bs[2 * l + 0]));
    float a1 = fmaf(Ws[4 * l + 2], h0, fmaf(Ws[4 * l + 3], h1, bs[2 * l + 1]));
    h0 = fast_tanh(a0);
    h1 = fast_tanh(a1);
  }
  out[i] = fmaf(Wout[0], h0, fmaf(Wout[1], h1, bout[0]));
}

// ---------- launcher ------------------------------------------------------

extern "C" void kernel_launch(void* const* d_in, const int* in_sizes, int n_in,
                              void* d_out, int out_size, void* d_ws, size_t ws_size,
                              hipStream_t stream) {
  const float* x    = (const float*)d_in[0];
  const float* Ws   = (const float*)d_in[1];
  const float* bs   = (const float*)d_in[2];
  const float* Wout = (const float*)d_in[3];
  const float* bout = (const float*)d_in[4];
  float* out = (float*)d_out;

  const int n = in_sizes[0] / 2;          // number of 2-D points
  const int numTiles = n / TILE_PTS;
  const int fullPts = numTiles * TILE_PTS;

  if (numTiles > 0) {
    int blocks = numTiles < MAX_BLOCKS ? numTiles : MAX_BLOCKS;
    int tpbk = (numTiles + blocks - 1) / blocks;
    mlp_tiles<<<blocks, TPB, 0, stream>>>(x, Ws, bs, Wout, bout, out,
                                          numTiles, tpbk);
  }
  const int rem = n - fullPts;
  if (rem > 0) {
    mlp_tail<<<(rem + TPB - 1) / TPB, TPB, 0, stream>>>(x, Ws, bs, Wout, bout,
                                                        out, fullPts, n);
  }
}